// PlanetoidGIN_51780125720797
// MI455X (gfx1250) — compile-verified
//
#include <hip/hip_runtime.h>
#include <hip/hip_bf16.h>

// GIN (3-layer) + global_add_pool, fp32 throughout.
// Edge scatter/gather is the bottleneck (L2-resident, ~5GB effective traffic);
// GEMMs use V_WMMA_F32_16X16X4_F32 so the matrix pipe handles the dense math
// at full fp32 precision.

#define DIM   128
#define NCLS  40
#define NGRAPH 64

typedef float v2f __attribute__((ext_vector_type(2)));
typedef float v8f __attribute__((ext_vector_type(8)));

// ---------------------------------------------------------------- zero fill
__global__ __launch_bounds__(256) void gin_zero4(float4* __restrict__ p, int n4) {
    int i = blockIdx.x * blockDim.x + threadIdx.x;
    if (i < n4) p[i] = make_float4(0.f, 0.f, 0.f, 0.f);
}

// ------------------------------------------------- edge scatter: agg += w*h[src]
// One wave per edge: lane L loads float4 (16B) -> 512B coalesced row read,
// then 4 f32 atomics into the destination row (served by L2 atomic units).
__global__ __launch_bounds__(256) void gin_scatter(
        const float* __restrict__ h,
        const int*   __restrict__ src,
        const int*   __restrict__ dst,
        const float* __restrict__ w,
        float*       __restrict__ agg,
        int nEdges) {
    long long gid = (long long)blockIdx.x * blockDim.x + threadIdx.x;
    int e    = (int)(gid >> 5);
    int lane = (int)(gid & 31);
    if (e >= nEdges) return;
    int   s  = src[e];
    int   d  = dst[e];
    float ww = w[e];
    const float4* hp = (const float4*)(h + (size_t)s * DIM);
    float4 v = hp[lane];
    float* ap = agg + (size_t)d * DIM + (size_t)lane * 4;
    atomicAdd(ap + 0, ww * v.x);
    atomicAdd(ap + 1, ww * v.y);
    atomicAdd(ap + 2, ww * v.z);
    atomicAdd(ap + 3, ww * v.w);
}

// ------------------------------------- hidden layer: out = act((h+agg)@W + b)
// Block = 256 threads = 8 waves; wave w computes the 16x16 tile at cols [16w,16w+16)
// of the 16-row strip at rows [16*blockIdx.x, ...). K looped in steps of 4 with
// V_WMMA_F32_16X16X4_F32. Loads are index-clamped so EXEC stays all-1s; the
// store path branches once on a wave-uniform full-tile test (no per-row cmpx).
template<int DO_RELU>
__global__ __launch_bounds__(256) void gin_gemm128(
        const float* __restrict__ h,
        const float* __restrict__ agg,
        const float* __restrict__ W,      // [DIM, DIM] row-major
        const float* __restrict__ bias,
        float*       __restrict__ out,    // [n, DIM]
        int nRows) {
    int wave  = threadIdx.x >> 5;            // 0..7 -> col tile
    int lane  = threadIdx.x & 31;
    int row0  = blockIdx.x * 16;
    int col0  = wave * 16;
    int mn    = lane & 15;                   // M index for A, N index for B/D
    int khalf = (lane >> 4) * 2;             // 0 or 2

    int rA = row0 + mn;
    if (rA > nRows - 1) rA = nRows - 1;      // clamp (keeps EXEC uniform)
    const float2* h2 = (const float2*)(h   + (size_t)rA * DIM);
    const float2* a2 = (const float2*)(agg + (size_t)rA * DIM);

    v8f acc = {};
    #pragma unroll 4
    for (int k = 0; k < DIM; k += 4) {
        int kk = k + khalf;
        float2 hv = h2[kk >> 1];
        float2 av = a2[kk >> 1];
        v2f a; a.x = hv.x + av.x; a.y = hv.y + av.y;
        v2f b; b.x = W[(size_t)kk * DIM + col0 + mn];
               b.y = W[(size_t)(kk + 1) * DIM + col0 + mn];
        acc = __builtin_amdgcn_wmma_f32_16x16x4_f32(
                  false, a, false, b, (short)0, acc, false, false);
    }

    float bv   = bias[col0 + mn];
    int   rOff = row0 + ((lane >> 4) ? 8 : 0);
    float* op  = out + (size_t)rOff * DIM + col0 + mn;

    if (row0 + 16 <= nRows) {                // uniform: common path, no masks
        #pragma unroll
        for (int r = 0; r < 8; ++r) {
            float v = acc[r] + bv;
            if (DO_RELU) v = fmaxf(v, 0.f);
            op[(size_t)r * DIM] = v;
        }
    } else {                                 // tail strip
        #pragma unroll
        for (int r = 0; r < 8; ++r) {
            if (rOff + r < nRows) {
                float v = acc[r] + bv;
                if (DO_RELU) v = fmaxf(v, 0.f);
                op[(size_t)r * DIM] = v;
            }
        }
    }
}

// ------------------------------ output layer: h3 = (h+agg)@W3 + b3  (C=40->48)
__global__ __launch_bounds__(96) void gin_gemm_out(
        const float* __restrict__ h,
        const float* __restrict__ agg,
        const float* __restrict__ W,      // [DIM, NCLS] row-major
        const float* __restrict__ bias,
        float*       __restrict__ out,    // [n, NCLS]
        int nRows) {
    int wave  = threadIdx.x >> 5;            // 0..2 -> col tile
    int lane  = threadIdx.x & 31;
    int row0  = blockIdx.x * 16;
    int col0  = wave * 16;
    int mn    = lane & 15;
    int khalf = (lane >> 4) * 2;

    int col  = col0 + mn;
    int colC = (col < NCLS) ? col : (NCLS - 1);
    float cmask = (col < NCLS) ? 1.f : 0.f;

    int rA = row0 + mn;
    if (rA > nRows - 1) rA = nRows - 1;
    const float2* h2 = (const float2*)(h   + (size_t)rA * DIM);
    const float2* a2 = (const float2*)(agg + (size_t)rA * DIM);

    v8f acc = {};
    #pragma unroll 4
    for (int k = 0; k < DIM; k += 4) {
        int kk = k + khalf;
        float2 hv = h2[kk >> 1];
        float2 av = a2[kk >> 1];
        v2f a; a.x = hv.x + av.x; a.y = hv.y + av.y;
        v2f b; b.x = W[(size_t)kk * NCLS + colC] * cmask;
               b.y = W[(size_t)(kk + 1) * NCLS + colC] * cmask;
        acc = __builtin_amdgcn_wmma_f32_16x16x4_f32(
                  false, a, false, b, (short)0, acc, false, false);
    }

    float bv   = bias[colC];
    int   rOff = row0 + ((lane >> 4) ? 8 : 0);
    float* op  = out + (size_t)rOff * NCLS + col;

    if (row0 + 16 <= nRows) {
        if (col < NCLS) {                    // single divergence region
            #pragma unroll
            for (int r = 0; r < 8; ++r)
                op[(size_t)r * NCLS] = acc[r] + bv;
        }
    } else {
        if (col < NCLS) {
            #pragma unroll
            for (int r = 0; r < 8; ++r)
                if (rOff + r < nRows)
                    op[(size_t)r * NCLS] = acc[r] + bv;
        }
    }
}

// ------------------------------------------- pool: out[g,c] += h3[i,c], batch sorted
// Thread c accumulates a contiguous node chunk in a register; one atomic per
// (graph transition, class) -> ~16k atomics total instead of 4M.
__global__ __launch_bounds__(64) void gin_pool(
        const float* __restrict__ h3,
        const int*   __restrict__ batch,
        float*       __restrict__ out,    // [NGRAPH, NCLS], pre-zeroed
        int nRows, int chunk) {
    int c = threadIdx.x;
    if (c >= NCLS) return;
    int start = blockIdx.x * chunk;
    if (start >= nRows) return;
    int end = start + chunk; if (end > nRows) end = nRows;

    float acc = 0.f;
    int g = batch[start];
    for (int i = start; i < end; ++i) {
        int gi = batch[i];
        if (gi != g) {
            atomicAdd(&out[g * NCLS + c], acc);
            acc = 0.f; g = gi;
        }
        acc += h3[(size_t)i * NCLS + c];
    }
    atomicAdd(&out[g * NCLS + c], acc);
}

// ---------------------------------------------------------------- launcher
extern "C" void kernel_launch(void* const* d_in, const int* in_sizes, int n_in,
                              void* d_out, int out_size, void* d_ws, size_t ws_size,
                              hipStream_t stream) {
    const float* x     = (const float*)d_in[0];
    const int*   ei    = (const int*)  d_in[1];
    const float* ew    = (const float*)d_in[2];
    const int*   batch = (const int*)  d_in[3];
    const float* W1    = (const float*)d_in[4];
    const float* b1    = (const float*)d_in[5];
    const float* W2    = (const float*)d_in[6];
    const float* b2    = (const float*)d_in[7];
    const float* W3    = (const float*)d_in[8];
    const float* b3    = (const float*)d_in[9];
    float* out = (float*)d_out;

    const int n = in_sizes[0] / DIM;      // nodes
    const int E = in_sizes[2];            // edges
    const int* src = ei;
    const int* dst = ei + E;

    // Workspace: agg | h1 | h2 | h3   (~170 MB)
    size_t nd = (size_t)n * DIM;
    float* agg = (float*)d_ws;
    float* h1  = agg + nd;
    float* h2  = h1 + nd;
    float* h3  = h2 + nd;

    const int ZB = 256;
    int nd4      = (int)(nd / 4);
    dim3 gZero((nd4 + ZB - 1) / ZB);
    dim3 gScat((unsigned)(((long long)E * 32 + ZB - 1) / ZB));
    dim3 gGemm((n + 15) / 16);
    const int chunk = 512;
    dim3 gPool((n + chunk - 1) / chunk);

    // zero the output accumulator (NGRAPH*NCLS = 2560 floats, %4==0)
    gin_zero4<<<dim3((NGRAPH * NCLS / 4 + ZB - 1) / ZB), ZB, 0, stream>>>(
        (float4*)out, NGRAPH * NCLS / 4);

    // ---- layer 1
    gin_zero4<<<gZero, ZB, 0, stream>>>((float4*)agg, nd4);
    gin_scatter<<<gScat, ZB, 0, stream>>>(x, src, dst, ew, agg, E);
    gin_gemm128<1><<<gGemm, 256, 0, stream>>>(x, agg, W1, b1, h1, n);

    // ---- layer 2
    gin_zero4<<<gZero, ZB, 0, stream>>>((float4*)agg, nd4);
    gin_scatter<<<gScat, ZB, 0, stream>>>(h1, src, dst, ew, agg, E);
    gin_gemm128<1><<<gGemm, 256, 0, stream>>>(h1, agg, W2, b2, h2, n);

    // ---- layer 3 (no relu, C=40)
    gin_zero4<<<gZero, ZB, 0, stream>>>((float4*)agg, nd4);
    gin_scatter<<<gScat, ZB, 0, stream>>>(h2, src, dst, ew, agg, E);
    gin_gemm_out<<<gGemm, 96, 0, stream>>>(h2, agg, W3, b3, h3, n);

    // ---- global add pool
    gin_pool<<<gPool, 64, 0, stream>>>(h3, batch, out, n, chunk);
}